// CAO_54829552501085
// MI455X (gfx1250) — compile-verified
//
#include <hip/hip_runtime.h>
#include <hip/hip_bf16.h>

#define MDIM 12
#define NDIM 32
#define NN   1024   // n*n
#define NPAD 16

typedef __attribute__((ext_vector_type(2))) float v2f;
typedef __attribute__((ext_vector_type(8))) float v8f;

// ---------------------------------------------------------------------------
// Zero a small float buffer (pair_aff + norm slot).
// ---------------------------------------------------------------------------
__global__ void cao_zero(float* __restrict__ p, int nfl) {
    int t = blockIdx.x * blockDim.x + threadIdx.x;
    if (t < nfl) p[t] = 0.f;
}

// ---------------------------------------------------------------------------
// pair_aff[i,j] = vec(X[i,j])^T K[i,j] vec(X[i,j])  (column-major vec)
// grid = (144 pairs, 8 row-chunks of 128), block = 256 (8 waves)
// ---------------------------------------------------------------------------
__global__ __launch_bounds__(256) void cao_pair_aff(
    const float* __restrict__ K, const float* __restrict__ X,
    float* __restrict__ pair_aff)
{
    int ij = blockIdx.x;
    int i = ij / MDIM, j = ij % MDIM;
    if (i == j) return;

    __shared__ float svx[NN];
    const float* Xij = X + (size_t)ij * NN;     // (32,32) row-major
    for (int p = threadIdx.x; p < NN; p += 256) {
        int b = p >> 5, a = p & 31;             // p = b*32 + a -> Xij[a][b]
        svx[p] = Xij[a * 32 + b];
    }
    __syncthreads();

    const float* Kij = K + (size_t)ij * NN * NN;
    int row0 = blockIdx.y * 128;
    float acc = 0.f;
    for (int p = row0; p < row0 + 128; ++p) {
        float vxp = svx[p];
        const float* krow = Kij + (size_t)p * NN;
        for (int q = threadIdx.x; q < NN; q += 256)
            acc += vxp * krow[q] * svx[q];
    }
    __shared__ float red[8];
    for (int off = 16; off; off >>= 1) acc += __shfl_xor(acc, off, 32);
    if ((threadIdx.x & 31) == 0) red[threadIdx.x >> 5] = acc;
    __syncthreads();
    if (threadIdx.x == 0) {
        float s = 0.f;
        for (int w = 0; w < 8; ++w) s += red[w];
        atomicAdd(&pair_aff[ij], s);
    }
}

// ---------------------------------------------------------------------------
// norm = max over off-diagonal pair_aff
// ---------------------------------------------------------------------------
__global__ void cao_norm(const float* __restrict__ pair_aff, float* __restrict__ normp) {
    float mx = 0.f;
    for (int ij = threadIdx.x; ij < MDIM * MDIM; ij += 32) {
        int i = ij / MDIM, j = ij % MDIM;
        float v = (i == j) ? 0.f : pair_aff[ij];
        mx = fmaxf(mx, v);
    }
    for (int off = 16; off; off >>= 1) mx = fmaxf(mx, __shfl_xor(mx, off, 32));
    if (threadIdx.x == 0) *normp = mx;
}

// ---------------------------------------------------------------------------
// Build V (16 x 1024, zero-padded): rows 0..11 = vec(C[k]^T) with
// C[k] = X[i,k] @ X[k,j]; row 12 = vec(X[i,j]^T); rows 13..15 = 0.
// Also zeroes aff[0..15]. One block of 256.
// ---------------------------------------------------------------------------
__global__ __launch_bounds__(256) void cao_buildV(
    const float* __restrict__ X, int i, int j,
    float* __restrict__ V, float* __restrict__ aff)
{
    __shared__ float sA[NN], sB[NN];
    int tid = threadIdx.x;
    if (tid < 16) aff[tid] = 0.f;
    for (int p = tid; p < NN; p += 256) {
        // V[12][b*32+a] = Xij[a][b]  (p = b*32+a)
        V[12 * NN + p] = X[((size_t)i * MDIM + j) * NN + ((p & 31) * 32 + (p >> 5))];
        V[13 * NN + p] = 0.f;
        V[14 * NN + p] = 0.f;
        V[15 * NN + p] = 0.f;
    }
    for (int k = 0; k < MDIM; ++k) {
        __syncthreads();
        for (int p = tid; p < NN; p += 256) {
            sA[p] = X[((size_t)i * MDIM + k) * NN + p];   // X[i,k]
            sB[p] = X[((size_t)k * MDIM + j) * NN + p];   // X[k,j]
        }
        __syncthreads();
        for (int idx = tid; idx < NN; idx += 256) {
            int a = idx >> 5, b = idx & 31;
            float s = 0.f;
#pragma unroll 8
            for (int c = 0; c < 32; ++c) s += sA[a * 32 + c] * sB[c * 32 + b];
            V[k * NN + b * 32 + a] = s;                   // column-major vec
        }
    }
}

// ---------------------------------------------------------------------------
// aff[M] += sum_q (sum_{p in kchunk} V[M,p] * Kij[p,q]) * V[M,q]
// via V_WMMA_F32_16X16X4_F32.  grid = (8 col-chunks of 128, 16 k-chunks of 64),
// block = 256 = 8 waves; each wave owns a 16-column tile.
// A fragment (16x4 f32): lanes 0-15 hold M=0..15; VGPR0 = K{0 | 2}, VGPR1 = K{1 | 3}.
// B fragment (4x16 f32): lanes 0-15 hold N;       VGPR0 = K{0 | 2}, VGPR1 = K{1 | 3}.
// All hot-loop loads are immediate-offset from two per-lane base pointers.
// ---------------------------------------------------------------------------
__global__ __launch_bounds__(256) void cao_gemm_aff(
    const float* __restrict__ Kij,   // 1024 x 1024
    const float* __restrict__ V,     // 16 x 1024 (rows 13..15 zero)
    float* __restrict__ aff)         // 16 floats, atomicAdd
{
    const int tid  = threadIdx.x;
    const int wave = tid >> 5;
    const int lane = tid & 31;
    const int half = lane >> 4;       // 0: K={0,1}, 1: K={2,3}
    const int l16  = lane & 15;
    const int p0   = blockIdx.y * 64;                      // k-chunk start
    const int q    = blockIdx.x * 128 + wave * 16 + l16;   // global column

    // Per-lane base pointers; every load below is base + compile-time offset.
    const float* aBase = V   + (size_t)l16 * NN + p0 + 2 * half;        // 8B aligned
    const float* bBase = Kij + (size_t)(p0 + 2 * half) * NN + q;

    v8f acc = {};
#pragma unroll
    for (int ks = 0; ks < 16; ++ks) {
        v2f a = *(const v2f*)(aBase + ks * 4);             // V[l16][p+2h .. p+2h+1]
        v2f b;
        b.x = bBase[ks * 4 * NN];                          // Kij[p+2h  ][q]
        b.y = bBase[ks * 4 * NN + NN];                     // Kij[p+2h+1][q]
        if (ks < 14)
            __builtin_prefetch(bBase + (ks + 2) * 4 * NN, 0, 1);
        acc = __builtin_amdgcn_wmma_f32_16x16x4_f32(
                  false, a, false, b, (short)0, acc, false, false);
    }

    // Epilogue: W[M,q] * V[M,q], reduce across the 16 lanes of each half.
#pragma unroll
    for (int r = 0; r < 8; ++r) {
        int M = r + 8 * half;                      // C/D layout: VGPR r, half selects M/M+8
        float prod = acc[r] * V[(size_t)M * NN + q];
        for (int off = 1; off < 16; off <<= 1)
            prod += __shfl_xor(prod, off, 32);     // stays within the 16-lane half
        if (l16 == 0 && M < 13) atomicAdd(&aff[M], prod);
    }
}

// ---------------------------------------------------------------------------
// Scores + consistency + update of X[i,j], X[j,i]. One block of 256.
// ---------------------------------------------------------------------------
__global__ __launch_bounds__(256) void cao_finalize(
    float* __restrict__ Xc, const float* __restrict__ V,
    const float* __restrict__ aff, const float* __restrict__ normp,
    int i, int j, int use_con, float cval)
{
    __shared__ float D[13 * 13];
    __shared__ int sbest;
    int tid = threadIdx.x, wave = tid >> 5, lane = tid & 31;

    if (use_con) {
        for (int pr = wave; pr < 169; pr += 8) {
            int k = pr / 13, l = pr % 13;
            float s = 0.f;
            for (int p = lane; p < NN; p += 32)
                s += fabsf(V[k * NN + p] - V[l * NN + p]);
            for (int off = 16; off; off >>= 1) s += __shfl_xor(s, off, 32);
            if (lane == 0) D[pr] = s;
        }
    }
    __syncthreads();

    if (tid == 0) {
        float inv = 1.0f / (*normp);
        float denom = 2.0f * (float)NDIM * (float)MDIM;
        float aff_ori = aff[12] * inv;
        float score_ori;
        if (use_con) {
            float s = 0.f;
            for (int k = 0; k < MDIM; ++k) s += D[12 * 13 + k];
            float con_ori = 1.0f - s / denom;
            score_ori = aff_ori * (1.0f - cval) + con_ori * cval;
        } else {
            score_ori = aff_ori;
        }
        int best = -1;
        for (int k = 0; k < MDIM; ++k) {
            float sc = aff[k] * inv;
            if (use_con) {
                float s = 0.f;
                for (int l = 0; l < MDIM; ++l) s += D[k * 13 + l];
                float cc = 1.0f - s / denom;
                sc = sc * (1.0f - cval) + cc * cval;
            }
            if (sc > score_ori) best = k;   // last qualifying k wins
        }
        sbest = best;
    }
    __syncthreads();

    int best = sbest;
    int src = (best >= 0) ? best : 12;      // row 12 holds vec(Xij^T)
    for (int idx = tid; idx < NN; idx += 256) {
        int a = idx >> 5, b = idx & 31;
        float val = V[src * NN + b * 32 + a];            // C[src][a][b] (or Xij[a][b])
        Xc[((size_t)i * MDIM + j) * NN + a * 32 + b] = val;
        Xc[((size_t)j * MDIM + i) * NN + b * 32 + a] = val;  // transpose
    }
}

// ---------------------------------------------------------------------------
extern "C" void kernel_launch(void* const* d_in, const int* in_sizes, int n_in,
                              void* d_out, int out_size, void* d_ws, size_t ws_size,
                              hipStream_t stream) {
    const float* K  = (const float*)d_in[0];
    const float* X0 = (const float*)d_in[1];
    float* Xc = (float*)d_out;               // working X == final output
    float* ws = (float*)d_ws;
    float* pair_aff = ws;                    // 144 floats
    float* normp    = ws + 144;              // 1 float
    float* V        = ws + 256;              // 16*1024 floats
    float* aff      = ws + 256 + NPAD * NN;  // 16 floats

    hipMemcpyAsync(Xc, X0, sizeof(float) * MDIM * MDIM * NN,
                   hipMemcpyDeviceToDevice, stream);

    float cval = 0.3f;
    for (int it = 0; it < 6; ++it) {
        int use_con = (it >= 2) ? 1 : 0;
        if (use_con) cval = fminf(cval * 1.1f, 1.0f);

        cao_zero<<<1, 256, 0, stream>>>(pair_aff, 145);
        cao_pair_aff<<<dim3(MDIM * MDIM, 8), 256, 0, stream>>>(K, Xc, pair_aff);
        cao_norm<<<1, 32, 0, stream>>>(pair_aff, normp);

        for (int i = 0; i < MDIM; ++i) {
            for (int j = i + 1; j < MDIM; ++j) {
                const float* Kij = K + ((size_t)i * MDIM + j) * NN * NN;
                cao_buildV<<<1, 256, 0, stream>>>(Xc, i, j, V, aff);
                cao_gemm_aff<<<dim3(8, 16), 256, 0, stream>>>(Kij, V, aff);
                cao_finalize<<<1, 256, 0, stream>>>(Xc, V, aff, normp, i, j,
                                                    use_con, cval);
            }
        }
    }
}